// HierarchicalAffinityAveraging_69587060129920
// MI455X (gfx1250) — compile-verified
//
#include <hip/hip_runtime.h>
#include <cstdint>
#include <cstddef>

// Problem constants (match reference setup_inputs: B=4, C=34, H=W=512, fp32)
#define HH   512
#define WW   512
#define CC   34
#define NEMB 16          // embedding channels (18..33)
#define HWP  (HH * WW)

#define TS   16          // output tile: 16x16 pixels
#define RS   24          // raw-emb staged region: TS + 2*4 (halo 1 for lvl0 + 3 for lvl1)
#define ES   22          // level-0 result region:  TS + 2*3
#define NTHREADS 256

__device__ __forceinline__ int iclamp(int v, int lo, int hi) {
    return v < lo ? lo : (v > hi ? hi : v);
}

// Per-pixel: softmax over 9 values + redundant L1 renorm (faithful to reference).
__device__ __forceinline__ void softmax9(float* aw) {
    float m = aw[0];
#pragma unroll
    for (int j = 1; j < 9; ++j) m = fmaxf(m, aw[j]);
    float s = 0.f;
#pragma unroll
    for (int j = 0; j < 9; ++j) { aw[j] = __expf(aw[j] - m); s += aw[j]; }
    float inv = 1.0f / s;
    float s2 = 0.f;
#pragma unroll
    for (int j = 0; j < 9; ++j) { aw[j] *= inv; s2 += fabsf(aw[j]); }
    float inv2 = 1.0f / fmaxf(s2, 1e-12f);
#pragma unroll
    for (int j = 0; j < 9; ++j) aw[j] *= inv2;
}

__global__ __launch_bounds__(NTHREADS)
void haa_fused_kernel(const float* __restrict__ in, float* __restrict__ out)
{
    // 36864 B, reused three times:
    //  1) raw emb, channel planes:   16 x 24x24 floats           (36864 B)
    //  2) level-0 result, px-major:  4 float4-groups x 22x22     (30976 B)
    //  3) output tile, ch planes:    16 x 16x16 floats           (16384 B)
    __shared__ __align__(16) float smem[NEMB * RS * RS];

    const int tid = threadIdx.x;
    const int b   = blockIdx.z;
    const int h0  = blockIdx.y * TS;
    const int w0  = blockIdx.x * TS;
    const float* src     = in  + (size_t)b * CC * HWP;
    float*       dstBase = out + (size_t)b * NEMB * HWP;

    // ---------------- Phase 0: async-stage raw embedding into LDS ----------
    // LDS layout: per-channel planes  smem[c*(RS*RS) + ry*RS + rx]
    if (w0 >= 4 && (w0 + TS + 4) <= WW) {
        // Interior in x: each 24-float row segment is contiguous & 16B aligned
        // (w0 multiple of 16 -> (w0-4)*4 is 16B aligned). 2304 = 9 * 256.
        for (int i = tid; i < NEMB * RS * (RS / 4); i += NTHREADS) {
            const int c   = i / (RS * (RS / 4));
            const int rem = i % (RS * (RS / 4));
            const int ry  = rem / (RS / 4);
            const int k   = rem % (RS / 4);
            const int gr  = iclamp(h0 - 4 + ry, 0, HH - 1);
            const uint32_t goff =
                (uint32_t)((((c + 18) * HWP) + gr * WW + (w0 - 4) + k * 4) * 4);
            const int loff = (c * (RS * RS) + ry * RS + k * 4) * 4;
            asm volatile("global_load_async_to_lds_b128 %0, %1, %2 offset:0"
                         :: "v"(loff), "v"(goff), "s"(src) : "memory");
        }
    } else {
        // x-border tiles: column clamping breaks contiguity -> per-element.
        // 9216 = 36 * 256 (all lanes active every iteration).
        for (int i = tid; i < NEMB * RS * RS; i += NTHREADS) {
            const int c  = i / (RS * RS);
            const int p  = i % (RS * RS);
            const int ry = p / RS;
            const int rx = p % RS;
            const int gr = iclamp(h0 - 4 + ry, 0, HH - 1);
            const int gc = iclamp(w0 - 4 + rx, 0, WW - 1);
            const uint32_t goff = (uint32_t)((((c + 18) * HWP) + gr * WW + gc) * 4);
            const int loff = i * 4;
            asm volatile("global_load_async_to_lds_b32 %0, %1, %2 offset:0"
                         :: "v"(loff), "v"(goff), "s"(src) : "memory");
        }
    }
    asm volatile("s_wait_asynccnt 0" ::: "memory");
    __syncthreads();

    // ---------------- Phase A: level 0 over extended 22x22 region ----------
    // Results stay in registers until all waves are done reading raw LDS.
    float acc[2][NEMB];
#pragma unroll
    for (int it = 0; it < 2; ++it) {
        const int p = tid + it * NTHREADS;
        if (p < ES * ES) {
            const int ey = p / ES, ex = p % ES;
            const int gh = h0 - 3 + ey, gw = w0 - 3 + ex;
            const int ghc = iclamp(gh, 0, HH - 1);
            const int gwc = iclamp(gw, 0, WW - 1);

            float aw[9];
#pragma unroll
            for (int j = 0; j < 9; ++j)
                aw[j] = src[(size_t)j * HWP + (size_t)ghc * WW + gwc];
            softmax9(aw);

            int r[3], q[3];
#pragma unroll
            for (int k = 0; k < 3; ++k) {
                r[k] = iclamp(gh + k - 1, 0, HH - 1) - (h0 - 4);
                q[k] = iclamp(gw + k - 1, 0, WW - 1) - (w0 - 4);
            }
#pragma unroll
            for (int c = 0; c < NEMB; ++c) acc[it][c] = 0.f;
#pragma unroll
            for (int dy = 0; dy < 3; ++dy) {
#pragma unroll
                for (int dx = 0; dx < 3; ++dx) {
                    const float wgt = aw[dy * 3 + dx];
                    const int base  = r[dy] * RS + q[dx];
#pragma unroll
                    for (int c = 0; c < NEMB; ++c)
                        acc[it][c] = fmaf(wgt, smem[c * (RS * RS) + base], acc[it][c]);
                }
            }
        }
    }
    __syncthreads();   // all raw-emb reads complete -> safe to repurpose LDS

    // Level-0 results, channel-packed: E0[g*(ES*ES) + pix] = channels 4g..4g+3
    float4* E0 = reinterpret_cast<float4*>(smem);
#pragma unroll
    for (int it = 0; it < 2; ++it) {
        const int p = tid + it * NTHREADS;
        if (p < ES * ES) {
#pragma unroll
            for (int g = 0; g < 4; ++g)
                E0[g * (ES * ES) + p] = make_float4(acc[it][4 * g + 0], acc[it][4 * g + 1],
                                                    acc[it][4 * g + 2], acc[it][4 * g + 3]);
        }
    }
    __syncthreads();

    // ---------------- Phase B: level 1 (dilation 3), one pixel per thread --
    float4 o[4];
    {
        const int ty = tid / TS, tx = tid % TS;
        const int gh = h0 + ty, gw = w0 + tx;

        float aw[9];
#pragma unroll
        for (int j = 0; j < 9; ++j)
            aw[j] = src[(size_t)(9 + j) * HWP + (size_t)gh * WW + gw];
        softmax9(aw);

        int r[3], q[3];
#pragma unroll
        for (int k = 0; k < 3; ++k) {
            r[k] = iclamp(gh + 3 * (k - 1), 0, HH - 1) - (h0 - 3);
            q[k] = iclamp(gw + 3 * (k - 1), 0, WW - 1) - (w0 - 3);
        }

#pragma unroll
        for (int g = 0; g < 4; ++g) o[g] = make_float4(0.f, 0.f, 0.f, 0.f);

#pragma unroll
        for (int dy = 0; dy < 3; ++dy) {
#pragma unroll
            for (int dx = 0; dx < 3; ++dx) {
                const float wgt = aw[dy * 3 + dx];
                const int p     = r[dy] * ES + q[dx];
#pragma unroll
                for (int g = 0; g < 4; ++g) {
                    const float4 v = E0[g * (ES * ES) + p];   // ds_load_b128
                    o[g].x = fmaf(wgt, v.x, o[g].x);
                    o[g].y = fmaf(wgt, v.y, o[g].y);
                    o[g].z = fmaf(wgt, v.z, o[g].z);
                    o[g].w = fmaf(wgt, v.w, o[g].w);
                }
            }
        }
    }
    __syncthreads();   // all E0 reads complete -> safe to repurpose LDS again

    // ---------------- Phase C: async-drain output tile via LDS -------------
    // Output tile in channel planes: smem[c*(TS*TS) + ty*TS + tx]
#pragma unroll
    for (int g = 0; g < 4; ++g) {
        smem[(4 * g + 0) * (TS * TS) + tid] = o[g].x;
        smem[(4 * g + 1) * (TS * TS) + tid] = o[g].y;
        smem[(4 * g + 2) * (TS * TS) + tid] = o[g].z;
        smem[(4 * g + 3) * (TS * TS) + tid] = o[g].w;
    }
    __syncthreads();   // (includes dscnt drain) LDS tile complete

    // 16 channels x 16 rows x 64B rows; each lane drains one 16B chunk.
    // 1024 chunks = 4 * 256 (all lanes active). Global runs are contiguous,
    // 64B-aligned (w0 multiple of 16) full-cacheline stores.
    for (int i = tid; i < NEMB * (TS * TS) / 4; i += NTHREADS) {
        const int c    = i >> 6;          // channel
        const int rem  = i & 63;          // 16B chunk within 16x16 plane
        const int row  = rem >> 2;
        const int col4 = (rem & 3) * 4;
        const int loff = (c * (TS * TS) + rem * 4) * 4;
        const uint32_t goff =
            (uint32_t)((c * HWP + (h0 + row) * WW + w0 + col4) * 4);
        asm volatile("global_store_async_from_lds_b128 %0, %1, %2 offset:0"
                     :: "v"(goff), "v"(loff), "s"(dstBase) : "memory");
    }
    // S_ENDPGM performs an implicit wait-idle, draining ASYNCcnt.
}

extern "C" void kernel_launch(void* const* d_in, const int* in_sizes, int n_in,
                              void* d_out, int out_size, void* d_ws, size_t ws_size,
                              hipStream_t stream)
{
    (void)n_in; (void)out_size; (void)d_ws; (void)ws_size;
    const float* x = (const float*)d_in[0];
    float* y = (float*)d_out;
    const int B = in_sizes[0] / (CC * HWP);       // = 4
    dim3 grid(WW / TS, HH / TS, B);               // 32 x 32 x 4 tiles
    haa_fused_kernel<<<grid, NTHREADS, 0, stream>>>(x, y);
}